// NodeBlock_17008070492484
// MI455X (gfx1250) — compile-verified
//
#include <hip/hip_runtime.h>

#define N_NODES 200000
#define N_EDGES 800000
#define D 128
#define TM 64           // nodes per MLP block = 4 M-tiles of 16
#define LDA 260         // 256 + 4 floats pad
#define LDH 132         // 128 + 4 floats pad
#define SCAN_CHUNK 1024
#define NBLK ((N_NODES + SCAN_CHUNK - 1) / SCAN_CHUNK)   // 196

typedef __attribute__((ext_vector_type(2))) float v2f;
typedef __attribute__((ext_vector_type(8))) float v8f;

// ---------------------------------------------------------------------------
// CSR build, step 1: zero per-node in-degree counters.
// ---------------------------------------------------------------------------
__global__ void zero_deg(int* __restrict__ deg) {
    int i = blockIdx.x * blockDim.x + threadIdx.x;
    if (i < N_NODES) deg[i] = 0;
}

// ---------------------------------------------------------------------------
// CSR build, step 2: histogram receivers. 800K int atomics (native
// GLOBAL_ATOMIC_ADD_U32) instead of 102M f32 atomics.
// ---------------------------------------------------------------------------
__global__ void count_deg(const int* __restrict__ edge_index,
                          int* __restrict__ deg) {
    int e = blockIdx.x * blockDim.x + threadIdx.x;   // grid sized exactly
    atomicAdd(&deg[edge_index[N_EDGES + e]], 1);     // edge_index[1][e]
}

// ---------------------------------------------------------------------------
// CSR build, step 3a: per-1024-chunk degree sums (tree reduction in LDS).
// ---------------------------------------------------------------------------
__global__ void block_sums(const int* __restrict__ deg, int* __restrict__ bsums) {
    __shared__ int sdata[SCAN_CHUNK];
    int tid = threadIdx.x;
    int i   = blockIdx.x * SCAN_CHUNK + tid;
    sdata[tid] = (i < N_NODES) ? deg[i] : 0;
    __syncthreads();
    for (int s = SCAN_CHUNK / 2; s > 0; s >>= 1) {
        if (tid < s) sdata[tid] += sdata[tid + s];
        __syncthreads();
    }
    if (tid == 0) bsums[blockIdx.x] = sdata[0];
}

// ---------------------------------------------------------------------------
// CSR build, step 3b: single-block exclusive scan of the 196 chunk sums.
// Also writes offsets[N_NODES] = total (= N_EDGES).
// ---------------------------------------------------------------------------
__global__ void scan_top(const int* __restrict__ bsums,
                         int* __restrict__ boffsets,
                         int* __restrict__ offsets) {
    __shared__ int buf[256];
    int tid = threadIdx.x;
    int v   = (tid < NBLK) ? bsums[tid] : 0;
    buf[tid] = v;
    __syncthreads();
    for (int off = 1; off < 256; off <<= 1) {
        int t = (tid >= off) ? buf[tid - off] : 0;
        __syncthreads();
        buf[tid] += t;
        __syncthreads();
    }
    if (tid < NBLK) boffsets[tid] = buf[tid] - v;     // exclusive
    if (tid == NBLK - 1) offsets[N_NODES] = buf[tid]; // grand total
}

// ---------------------------------------------------------------------------
// CSR build, step 3c: local exclusive scan per chunk + chunk offset.
// Writes both offsets[] (row pointers) and cursor[] (fill cursors).
// ---------------------------------------------------------------------------
__global__ void scan_apply(const int* __restrict__ deg,
                           const int* __restrict__ boffsets,
                           int* __restrict__ offsets,
                           int* __restrict__ cursor) {
    __shared__ int buf[SCAN_CHUNK];
    int tid = threadIdx.x;
    int i   = blockIdx.x * SCAN_CHUNK + tid;
    int v   = (i < N_NODES) ? deg[i] : 0;
    buf[tid] = v;
    __syncthreads();
    for (int off = 1; off < SCAN_CHUNK; off <<= 1) {
        int t = (tid >= off) ? buf[tid - off] : 0;
        __syncthreads();
        buf[tid] += t;
        __syncthreads();
    }
    int excl = buf[tid] - v + boffsets[blockIdx.x];
    if (i < N_NODES) { offsets[i] = excl; cursor[i] = excl; }
}

// ---------------------------------------------------------------------------
// CSR build, step 4: scatter edge ids into CSR adjacency (800K int atomics).
// ---------------------------------------------------------------------------
__global__ void fill_edges(const int* __restrict__ edge_index,
                           int* __restrict__ cursor,
                           int* __restrict__ edge_ids) {
    int e = blockIdx.x * blockDim.x + threadIdx.x;   // grid sized exactly
    int r = edge_index[N_EDGES + e];
    int pos = atomicAdd(&cursor[r], 1);
    edge_ids[pos] = e;
}

// ---------------------------------------------------------------------------
// Fused gather + 2-layer MLP using V_WMMA_F32_16X16X4_F32.
// Block = 256 threads = 8 wave32; block owns 64 nodes (4 M-tiles).
// Staging: wave w gathers rows [8w, 8w+8): x row -> sA[:,0:128],
// CSR gather-sum of edge_attr rows -> sA[:,128:256). One wave reads a whole
// 512B edge row coalesced (lane = 4 features) -- no atomics anywhere.
// GEMM: wave w owns output columns [16w,16w+16); each B fragment (from W1/W2,
// L2-resident) feeds 4 WMMAs (one per M-tile), so the accumulator chains have
// dependency distance 4 and weight traffic is amortized 4x.
//
// f32 WMMA fragment layouts (ISA 7.12.2):
//   A 16x4 : lanes 0-15 -> M=lane,  VGPRs = K0,K1 ; lanes 16-31 -> K2,K3
//   B 4x16 : lane holds N=lane%16 ; VGPRs = K{0,1} (lanes<16) / K{2,3}
//   C/D    : VGPR i, lanes 0-15 -> (M=i,N=lane); lanes 16-31 -> (M=i+8,N-16)
// ---------------------------------------------------------------------------
__global__ __launch_bounds__(256) void node_mlp(
        const float*  __restrict__ x,
        const float4* __restrict__ edge_attr4,
        const int*    __restrict__ offsets,
        const int*    __restrict__ edge_ids,
        const float*  __restrict__ W1, const float* __restrict__ b1,
        const float*  __restrict__ W2, const float* __restrict__ b2,
        float* __restrict__ out) {
    __shared__ float sA[TM * LDA];   // 64 x 256 : [x | gathered agg]
    __shared__ float sH[TM * LDH];   // 64 x 128 : hidden activations

    const int node0 = blockIdx.x * TM;
    const int tid   = threadIdx.x;
    const int wave  = tid >> 5;
    const int lane  = tid & 31;

    // ---- stage: x rows + CSR gather of agg rows ----------------------------
    for (int r8 = 0; r8 < 8; ++r8) {
        const int row = wave * 8 + r8;
        const int n   = node0 + row;
        float4 xv = ((const float4*)x)[(size_t)n * (D / 4) + lane];
        *(float4*)&sA[row * LDA + lane * 4] = xv;

        float4 acc = make_float4(0.f, 0.f, 0.f, 0.f);
        const int e0 = offsets[n], e1 = offsets[n + 1];
        for (int e = e0; e < e1; ++e) {
            const int id = edge_ids[e];
            float4 ev = edge_attr4[(size_t)id * (D / 4) + lane];
            acc.x += ev.x; acc.y += ev.y; acc.z += ev.z; acc.w += ev.w;
        }
        *(float4*)&sA[row * LDA + D + lane * 4] = acc;
    }
    __syncthreads();

    const int n0    = wave * 16;          // this wave's N-tile
    const int mrow  = lane & 15;          // A-fragment row within M-tile
    const int khalf = (lane >> 4) << 1;   // 0 (lanes 0-15) or 2 (lanes 16-31)
    const int ncol  = n0 + mrow;          // this lane's output column
    const int mbase = (lane >> 4) << 3;   // C/D: M = vgpr_idx + mbase

    // ---- GEMM1: h = relu([x|agg] @ W1 + b1), K = 256 -----------------------
    v8f c1[4] = {};
    #pragma unroll 2
    for (int k0 = 0; k0 < 2 * D; k0 += 4) {
        v2f b;
        b.x = W1[(size_t)(k0 + khalf)     * D + ncol];
        b.y = W1[(size_t)(k0 + khalf + 1) * D + ncol];
        #pragma unroll
        for (int m = 0; m < 4; ++m) {
            v2f a = *(const v2f*)&sA[(m * 16 + mrow) * LDA + k0 + khalf];
            c1[m] = __builtin_amdgcn_wmma_f32_16x16x4_f32(
                        false, a, false, b, (short)0, c1[m], false, false);
        }
    }
    const float bias1 = b1[ncol];
    #pragma unroll
    for (int m = 0; m < 4; ++m)
        #pragma unroll
        for (int i = 0; i < 8; ++i) {
            float h = c1[m][i] + bias1;
            sH[(m * 16 + i + mbase) * LDH + ncol] = fmaxf(h, 0.f);
        }
    __syncthreads();

    // ---- GEMM2: out = h @ W2 + b2, K = 128 ---------------------------------
    v8f c2[4] = {};
    #pragma unroll 2
    for (int k0 = 0; k0 < D; k0 += 4) {
        v2f b;
        b.x = W2[(size_t)(k0 + khalf)     * D + ncol];
        b.y = W2[(size_t)(k0 + khalf + 1) * D + ncol];
        #pragma unroll
        for (int m = 0; m < 4; ++m) {
            v2f a = *(const v2f*)&sH[(m * 16 + mrow) * LDH + k0 + khalf];
            c2[m] = __builtin_amdgcn_wmma_f32_16x16x4_f32(
                        false, a, false, b, (short)0, c2[m], false, false);
        }
    }
    const float bias2 = b2[ncol];
    #pragma unroll
    for (int m = 0; m < 4; ++m)
        #pragma unroll
        for (int i = 0; i < 8; ++i)
            out[(size_t)(node0 + m * 16 + i + mbase) * D + ncol] = c2[m][i] + bias2;
}

// ---------------------------------------------------------------------------
// Host entry. Inputs: x, edge_attr, edge_index, W1, b1, W2, b2.
// Workspace (all int32, ~5.6 MB): deg | offsets(N+1) | cursor | bsums |
// boffsets | edge_ids. Every word is rewritten each call.
// ---------------------------------------------------------------------------
extern "C" void kernel_launch(void* const* d_in, const int* in_sizes, int n_in,
                              void* d_out, int out_size, void* d_ws, size_t ws_size,
                              hipStream_t stream) {
    const float* x         = (const float*)d_in[0];
    const float* edge_attr = (const float*)d_in[1];
    const int*   edge_idx  = (const int*)  d_in[2];
    const float* W1        = (const float*)d_in[3];
    const float* b1        = (const float*)d_in[4];
    const float* W2        = (const float*)d_in[5];
    const float* b2        = (const float*)d_in[6];
    float*       out       = (float*)d_out;

    int* wsI      = (int*)d_ws;
    int* deg      = wsI;                         // N_NODES
    int* offsets  = deg      + N_NODES;          // N_NODES + 1
    int* cursor   = offsets  + N_NODES + 1;      // N_NODES
    int* bsums    = cursor   + N_NODES;          // NBLK
    int* boffsets = bsums    + NBLK;             // NBLK
    int* edge_ids = boffsets + NBLK;             // N_EDGES

    // CSR build
    zero_deg  <<<(N_NODES + 255) / 256, 256, 0, stream>>>(deg);
    count_deg <<<N_EDGES / 256,        256, 0, stream>>>(edge_idx, deg);
    block_sums<<<NBLK, SCAN_CHUNK,          0, stream>>>(deg, bsums);
    scan_top  <<<1, 256,                    0, stream>>>(bsums, boffsets, offsets);
    scan_apply<<<NBLK, SCAN_CHUNK,          0, stream>>>(deg, boffsets, offsets, cursor);
    fill_edges<<<N_EDGES / 256,        256, 0, stream>>>(edge_idx, cursor, edge_ids);

    // Fused gather + MLP: 200000 / 64 = 3125 blocks
    node_mlp<<<N_NODES / TM, 256, 0, stream>>>(
        x, (const float4*)edge_attr, offsets, edge_ids, W1, b1, W2, b2, out);
}